// Enc_state_9929964388905
// MI455X (gfx1250) — compile-verified
//
#include <hip/hip_runtime.h>
#include <math.h>

typedef __attribute__((ext_vector_type(2))) float v2f;
typedef __attribute__((ext_vector_type(4))) float v4f;
typedef __attribute__((ext_vector_type(8))) float v8f;

#define S_  8
#define B_  16
#define NP  1024
#define DF  64
#define KC  8
#define HH  256
#define LSTR 260   // padded row stride (floats): 16B-aligned rows, bank stride 4,
                   // store halves land on disjoint banks (8*260%64 == 32)

// ---------------------------------------------------------------------------
// Kernel A: mup[row][h] = b1[h] + sum_d mu[row][d] * W1[(DF+d)*HH + h]
// row in [0, S*B*K). Coalesced on W1/h; mu row broadcast within each block.
// ---------------------------------------------------------------------------
__global__ __launch_bounds__(256)
void mu_part_kernel(const float* __restrict__ mu,
                    const float* __restrict__ W1,
                    const float* __restrict__ b1,
                    float* __restrict__ mup) {
    const int idx = blockIdx.x * 256 + threadIdx.x;   // S*B*K*H total
    const int h   = idx & (HH - 1);
    const int row = idx >> 8;
    const float* __restrict__ murow = mu + (size_t)row * DF;
    const float* __restrict__ w     = W1 + (size_t)DF * HH + h;
    float acc = b1[h];
#pragma unroll 8
    for (int d = 0; d < DF; ++d)
        acc = fmaf(murow[d], w[(size_t)d * HH], acc);
    mup[idx] = acc;
}

// ---------------------------------------------------------------------------
// Kernel B: one workgroup (4 waves / 128 thr) per (s,b, 16-row n-tile).
// Phase 1: ob_part tile (16 x 256) via v_wmma_f32_16x16x4_f32 -> LDS.
// Phase 2: logits[k][n] = sum_h relu(obp+mup)*W2 (float4 LDS); softmax; sample.
// ---------------------------------------------------------------------------
__global__ __launch_bounds__(128)
void enc_main_kernel(const float* __restrict__ ob,
                     const float* __restrict__ W1,
                     const float* __restrict__ W2,
                     const float* __restrict__ b2p,
                     const float* __restrict__ mup,
                     float* __restrict__ qout,
                     float* __restrict__ sout) {
    __shared__ __align__(16) float s_obp[16 * LSTR];   // ob_part tile, padded
    __shared__ __align__(16) float s_mup[KC * LSTR];   // mu_part(+b1), padded
    __shared__ __align__(16) float s_w2[HH];
    __shared__ float s_logit[16 * KC];

    const int tid  = threadIdx.x;
    const int tile = blockIdx.x;          // 0 .. S*B*(N/16)-1
    const int nt   = tile & 63;           // n-tile within (s,b)
    const int sb   = tile >> 6;           // s*B + b
    const int n0   = nt * 16;

    // Stage mu_part rows (padded stride) + W2 into LDS.
    const float* __restrict__ mup_sb = mup + (size_t)sb * (KC * HH);
    for (int i = tid; i < KC * HH; i += 128) {
        const int k = i >> 8, h = i & (HH - 1);
        s_mup[k * LSTR + h] = mup_sb[i];
    }
    for (int i = tid; i < HH; i += 128) s_w2[i] = W2[i];

    // ---- Phase 1: WMMA f32 16x16x4, 16 k-steps over D=64 ----
    const int lane = tid & 31;
    const int wave = tid >> 5;
    const int half = lane >> 4;           // K sub-pair select (A/B layout)
    const int lm   = lane & 15;           // M (A) / N (B) index

    const float* __restrict__ obrow =
        ob + (((size_t)sb * NP) + n0 + lm) * DF;
    const int hbase = wave * 64;          // this wave's 64 H-columns

    v8f c0 = {}, c1 = {}, c2 = {}, c3 = {};
#pragma unroll
    for (int kk = 0; kk < 16; ++kk) {
        const int d = kk * 4 + half * 2;
        v2f a;
        a.x = obrow[d];
        a.y = obrow[d + 1];
        const float* __restrict__ w0 = W1 + (size_t)d * HH;
        v2f b;
        b.x = w0[hbase +  0 + lm];  b.y = w0[HH + hbase +  0 + lm];
        c0 = __builtin_amdgcn_wmma_f32_16x16x4_f32(false, a, false, b,
                                                   (short)0, c0, false, false);
        b.x = w0[hbase + 16 + lm];  b.y = w0[HH + hbase + 16 + lm];
        c1 = __builtin_amdgcn_wmma_f32_16x16x4_f32(false, a, false, b,
                                                   (short)0, c1, false, false);
        b.x = w0[hbase + 32 + lm];  b.y = w0[HH + hbase + 32 + lm];
        c2 = __builtin_amdgcn_wmma_f32_16x16x4_f32(false, a, false, b,
                                                   (short)0, c2, false, false);
        b.x = w0[hbase + 48 + lm];  b.y = w0[HH + hbase + 48 + lm];
        c3 = __builtin_amdgcn_wmma_f32_16x16x4_f32(false, a, false, b,
                                                   (short)0, c3, false, false);
    }

    // D layout: VGPR r holds M = half*8 + r, N = lm (within each column tile)
#pragma unroll
    for (int r = 0; r < 8; ++r) {
        const int nrow = half * 8 + r;
        s_obp[nrow * LSTR + hbase +  0 + lm] = c0[r];
        s_obp[nrow * LSTR + hbase + 16 + lm] = c1[r];
        s_obp[nrow * LSTR + hbase + 32 + lm] = c2[r];
        s_obp[nrow * LSTR + hbase + 48 + lm] = c3[r];
    }
    __syncthreads();

    // ---- Phase 2: fused relu-dot over H, float4 LDS reads ----
    const float b2 = b2p[0];
    const int k = tid >> 4;               // 0..7
    const int n = tid & 15;               // 0..15
    {
        const v4f* __restrict__ op = (const v4f*)(s_obp + n * LSTR);
        const v4f* __restrict__ mp = (const v4f*)(s_mup + k * LSTR);
        const v4f* __restrict__ wp = (const v4f*)s_w2;
        v4f acc4 = {};
#pragma unroll 4
        for (int h4 = 0; h4 < HH / 4; ++h4) {
            v4f v = op[h4] + mp[h4];                 // b1 already folded in
            v4f z = {};
            v = __builtin_elementwise_max(v, z);     // relu
            acc4 += v * wp[h4];
        }
        s_logit[n * KC + k] = acc4.x + acc4.y + acc4.z + acc4.w + b2;
    }
    __syncthreads();

    // ---- Softmax over K + deterministic categorical sample ----
    if (tid < 16) {
        const int ng = n0 + tid;
        float lg[KC];
        float mx = -INFINITY;
#pragma unroll
        for (int j = 0; j < KC; ++j) {
            lg[j] = s_logit[tid * KC + j];
            mx = fmaxf(mx, lg[j]);
        }
        float sum = 0.f;
#pragma unroll
        for (int j = 0; j < KC; ++j) { lg[j] = __expf(lg[j] - mx); sum += lg[j]; }
        const float inv = 1.f / sum;

        const size_t base = ((size_t)sb * NP + ng) * KC;
        float p[KC];
#pragma unroll
        for (int j = 0; j < KC; ++j) { p[j] = lg[j] * inv; qout[base + j] = p[j]; }

        // Deterministic hash -> uniform in [0,1), then CDF walk.
        unsigned x = (unsigned)(sb * NP + ng) * 2654435761u + 0x9E3779B9u;
        x ^= x >> 16; x *= 0x7feb352du;
        x ^= x >> 15; x *= 0x846ca68bu;
        x ^= x >> 16;
        const float u = (float)(x >> 8) * (1.0f / 16777216.0f);
        int idx = KC - 1;
        float cdf = 0.f;
#pragma unroll
        for (int j = 0; j < KC; ++j) {
            cdf += p[j];
            if (u < cdf) { idx = j; break; }
        }
#pragma unroll
        for (int j = 0; j < KC; ++j)
            sout[base + j] = (j == idx) ? 1.0f : 0.0f;
    }
}

// ---------------------------------------------------------------------------
extern "C" void kernel_launch(void* const* d_in, const int* in_sizes, int n_in,
                              void* d_out, int out_size, void* d_ws, size_t ws_size,
                              hipStream_t stream) {
    (void)in_sizes; (void)n_in; (void)out_size; (void)ws_size;
    const float* ob = (const float*)d_in[0];
    const float* mu = (const float*)d_in[1];
    const float* W1 = (const float*)d_in[2];
    const float* b1 = (const float*)d_in[3];
    const float* W2 = (const float*)d_in[4];
    const float* b2 = (const float*)d_in[5];

    float* mup  = (float*)d_ws;                       // S*B*K*H floats = 1 MB
    float* qout = (float*)d_out;                      // (S,B,N,K) probs
    float* sout = qout + (size_t)S_ * B_ * NP * KC;   // (S,B,N,K) one-hot

    mu_part_kernel<<<(S_ * B_ * KC * HH) / 256, 256, 0, stream>>>(mu, W1, b1, mup);
    enc_main_kernel<<<S_ * B_ * (NP / 16), 128, 0, stream>>>(ob, W1, W2, b2,
                                                             mup, qout, sout);
}